// HashEncodingEnsemble_12266426597922
// MI455X (gfx1250) — compile-verified
//
#include <hip/hip_runtime.h>
#include <stdint.h>

// ---- problem constants (match reference) ----
#define NLEV   16
#define NTAB   16
#define LOG2T  19
#define TSIZE  (1u << LOG2T)
#define TMASK  (TSIZE - 1u)
#define PRIME1 2654435761u
#define PRIME2 805459861u
#define PER_LEVEL_SCALE 1.4472692012786865

typedef __attribute__((ext_vector_type(2))) float v2f;
typedef __attribute__((ext_vector_type(8))) float v8f;

#define NB_MAIN 1024   // blocks of main kernel (must be multiple of 4 for WMMA reduce)

// -----------------------------------------------------------------------------
// Main gather + interpolate + blend kernel.
// Thread layout: tid&15 = table k, tid>>4 = point slot (16 points per block per
// grid-stride step). Each thread keeps 32 f32 accumulators (one per output
// feature of its table) in registers; per point per level it issues 8
// independent global_load_b64 gathers (features are float2-contiguous).
// Hash/weight math is recomputed per table-lane (VALU is free vs. the gather).
// Block epilogue: deterministic LDS tree reduction over the 16 groups, then
// either write a 512-float partial to d_ws or (fallback) global atomicAdd.
// -----------------------------------------------------------------------------
__global__ __launch_bounds__(256) void hash_blend_kernel(
    const float* __restrict__ xin,      // N x 3
    const float* __restrict__ code,     // N x 16
    const float* __restrict__ tables,   // 16 x 16 x 2^19 x 2
    float* __restrict__ partials,       // NB x 512 (if useWs)
    float* __restrict__ out_atomic,     // 512 (fallback)
    int N, int pointStride, int useWs)
{
  const int tid  = threadIdx.x;
  const int k    = tid & 15;
  const int slot = tid >> 4;
  const float* __restrict__ tk =
      tables + (size_t)k * ((size_t)NLEV * (size_t)TSIZE * 2u);

  // Per-level resolution: 16 * scale^l computed in double, cast to f32
  // (matches the reference's double pow -> f32 promotion to last-ulp).
  float resf[NLEV];
  {
    double r = 16.0;
#pragma unroll
    for (int l = 0; l < NLEV; ++l) { resf[l] = (float)r; r *= PER_LEVEL_SCALE; }
  }

  float acc[2 * NLEV];
#pragma unroll
  for (int d = 0; d < 2 * NLEV; ++d) acc[d] = 0.0f;

  for (int n = blockIdx.x * 16 + slot; n < N; n += pointStride) {
    const float px = xin[3 * (size_t)n + 0];
    const float py = xin[3 * (size_t)n + 1];
    const float pz = xin[3 * (size_t)n + 2];
    const float cd = code[(size_t)n * NTAB + k];

#pragma unroll
    for (int l = 0; l < NLEV; ++l) {
      const float res = resf[l];
      const float posx = px * res, posy = py * res, posz = pz * res;
      const float f0x = floorf(posx), f0y = floorf(posy), f0z = floorf(posz);
      const float fx = posx - f0x, fy = posy - f0y, fz = posz - f0z;
      const uint32_t ix = (uint32_t)f0x, iy = (uint32_t)f0y, iz = (uint32_t)f0z;
      // prime0 == 1, so the x term is just the coordinate
      const uint32_t xs0 = ix,          xs1 = ix + 1u;
      const uint32_t ys0 = iy * PRIME1, ys1 = ys0 + PRIME1;
      const uint32_t zs0 = iz * PRIME2, zs1 = zs0 + PRIME2;
      const float wx0 = 1.0f - fx, wy0 = 1.0f - fy, wz0 = 1.0f - fz;
      const float2* __restrict__ tl =
          (const float2*)(tk + (size_t)l * (size_t)(TSIZE * 2u));
      float e0 = 0.0f, e1 = 0.0f;
#pragma unroll
      for (int c = 0; c < 8; ++c) {            // corner = (c>>2, c>>1, c) bits
        const uint32_t hx = ((c >> 2) & 1) ? xs1 : xs0;
        const uint32_t hy = ((c >> 1) & 1) ? ys1 : ys0;
        const uint32_t hz = (c & 1) ? zs1 : zs0;
        const uint32_t h  = (hx ^ hy ^ hz) & TMASK;
        const float w = (((c >> 2) & 1) ? fx : wx0) *
                        (((c >> 1) & 1) ? fy : wy0) *
                        ((c & 1) ? fz : wz0);
        const float2 f = tl[h];                // global_load_b64, random
        e0 = fmaf(w, f.x, e0);
        e1 = fmaf(w, f.y, e1);
      }
      acc[2 * l + 0] = fmaf(cd, e0, acc[2 * l + 0]);
      acc[2 * l + 1] = fmaf(cd, e1, acc[2 * l + 1]);
    }
  }

  // ---- deterministic block reduction over the 16 point-slots per table ----
  __shared__ float lbuf[256][33];              // +1 pad to dodge bank conflicts
#pragma unroll
  for (int d = 0; d < 32; ++d) lbuf[tid][d] = acc[d];
  __syncthreads();

  for (int idx = tid; idx < 512; idx += 256) { // idx = k*32 + d
    const int kk = idx >> 5;
    const int d  = idx & 31;
    float s = 0.0f;
#pragma unroll
    for (int g = 0; g < 16; ++g) s += lbuf[kk + 16 * g][d];
    if (useWs) partials[(size_t)blockIdx.x * 512 + idx] = s;
    else       atomicAdd(&out_atomic[idx], s);
  }
}

// -----------------------------------------------------------------------------
// Cross-block reduction via V_WMMA_F32_16X16X4_F32 with an all-ones B matrix:
// D[m,n] += sum_k A[m,k] * 1  -> every column of D holds the K-dim partial sum.
// A layout (ISA 7.12.2, 32-bit A 16x4): lanes 0-15 = rows M, VGPR0 = K0 (lo
// half) / K2 (hi half), VGPR1 = K1 / K3. One wave owns 16 of the 512 outputs
// and sweeps all NB_MAIN block-partials 4 at a time, accumulating in C (f32,
// RNE — exact same arithmetic class as scalar adds).
// D extraction at N=0: lane 0 holds rows 0-7 in c[0..7], lane 16 rows 8-15.
// -----------------------------------------------------------------------------
__global__ __launch_bounds__(1024) void reduce_wmma_kernel(
    const float* __restrict__ partials, float* __restrict__ out, int nb)
{
  const int wave = threadIdx.x >> 5;   // 0..31, one 16-output chunk per wave
  const int lane = threadIdx.x & 31;
  const int base = wave * 16;
  const int row  = lane & 15;
  const int koff = (lane >> 4) << 1;   // 0 for lanes 0-15, 2 for lanes 16-31

  v8f c = {};
  v2f bones; bones.x = 1.0f; bones.y = 1.0f;

  for (int b = 0; b < nb; b += 4) {
    v2f a;
    a.x = partials[(size_t)(b + koff + 0) * 512 + base + row];
    a.y = partials[(size_t)(b + koff + 1) * 512 + base + row];
    c = __builtin_amdgcn_wmma_f32_16x16x4_f32(
        /*neg_a=*/false, a, /*neg_b=*/false, bones,
        /*c_mod=*/(short)0, c, /*reuse_a=*/false, /*reuse_b=*/false);
  }

  if (lane == 0) {
#pragma unroll
    for (int j = 0; j < 8; ++j) out[base + j] = c[j];
  } else if (lane == 16) {
#pragma unroll
    for (int j = 0; j < 8; ++j) out[base + 8 + j] = c[j];
  }
}

__global__ void zero512_kernel(float* __restrict__ out)
{
  if (threadIdx.x < 512) out[threadIdx.x] = 0.0f;
}

extern "C" void kernel_launch(void* const* d_in, const int* in_sizes, int n_in,
                              void* d_out, int out_size, void* d_ws, size_t ws_size,
                              hipStream_t stream)
{
  const float* xin    = (const float*)d_in[0];   // (N,3)  f32
  const float* code   = (const float*)d_in[1];   // (N,16) f32
  const float* tables = (const float*)d_in[2];   // (16,16,2^19,2) f32
  float* out = (float*)d_out;                    // (16,32) f32

  const int N = in_sizes[0] / 3;
  const int pointStride = NB_MAIN * 16;
  const size_t need = (size_t)NB_MAIN * 512 * sizeof(float);

  if (d_ws != nullptr && ws_size >= need) {
    // Deterministic path: per-block partials -> WMMA ones-matmul reduction.
    hash_blend_kernel<<<NB_MAIN, 256, 0, stream>>>(
        xin, code, tables, (float*)d_ws, out, N, pointStride, /*useWs=*/1);
    reduce_wmma_kernel<<<1, 1024, 0, stream>>>((const float*)d_ws, out, NB_MAIN);
  } else {
    // Fallback: zero output then global atomic accumulation.
    zero512_kernel<<<1, 512, 0, stream>>>(out);
    hash_blend_kernel<<<NB_MAIN, 256, 0, stream>>>(
        xin, code, tables, nullptr, out, N, pointStride, /*useWs=*/0);
  }
}